// NeighborhoodAttention_23536420782227
// MI455X (gfx1250) — compile-verified
//
#include <hip/hip_runtime.h>
#include <hip/hip_bf16.h>

typedef __attribute__((ext_vector_type(2))) float v2f;
typedef __attribute__((ext_vector_type(4))) float v4f;
typedef __attribute__((ext_vector_type(8))) float v8f;
typedef __attribute__((ext_vector_type(4))) unsigned int u32x4;
typedef __attribute__((ext_vector_type(4))) int i32x4;
typedef __attribute__((ext_vector_type(8))) int i32x8;

#define NH 8
#define HD 32
#define HH 56
#define WW2 56
#define HWN 3136          // 56*56
#define KPOS 154          // 7*22 key patch
#define KPAD 160          // padded GEMM extent

#if __has_builtin(__builtin_amdgcn_tensor_load_to_lds)
#define HAVE_TDM 1
#else
#define HAVE_TDM 0
#endif

__device__ __forceinline__ int iclamp(int v, int lo, int hi) {
    return v < lo ? lo : (v > hi ? hi : v);
}

__device__ __forceinline__ v8f wmma_f32(v2f a, v2f b, v8f c) {
    // D = A(16x4,f32) * B(4x16,f32) + C(16x16,f32)
    return __builtin_amdgcn_wmma_f32_16x16x4_f32(
        false, a, false, b, (short)0, c, false, false);
}

#if HAVE_TDM
// ---------------------------------------------------------------------------
// Issue one Tensor-Data-Mover load: up-to-3D tile of f32 into LDS,
// contiguous in LDS (x fastest, then y, then z). OOB reads return zero.
//   td0/td1/td2  : remaining tensor extents from the tile origin (elements)
//   tile0/1/2    : tile dims (elements); tile2==0 -> 2D
//   s0 / s1      : element strides between y-lines / z-planes
// ---------------------------------------------------------------------------
__device__ __forceinline__ void tdm_load3d(unsigned int ldsOff, const float* gptr,
                                           unsigned int td0, unsigned int td1,
                                           unsigned int td2,
                                           unsigned int tile0, unsigned int tile1,
                                           unsigned int tile2,
                                           unsigned long long s0, unsigned long long s1)
{
    unsigned long long ga = (unsigned long long)(uintptr_t)gptr;
    u32x4 g0;
    g0[0] = 1u;                                   // count=1 (valid, user mode)
    g0[1] = ldsOff;                               // lds_addr (bytes)
    g0[2] = (unsigned int)ga;                     // global_addr[31:0]
    g0[3] = (unsigned int)(ga >> 32) | (2u << 30);// global_addr[56:32] | type=2
    i32x8 g1;
    g1[0] = (int)(2u << 16);                      // wg_mask=0, data_size=2 (4B)
    g1[1] = (int)((td0 & 0xFFFFu) << 16);         // abar=0 | tensor_dim0 lo
    g1[2] = (int)((td0 >> 16) | ((td1 & 0xFFFFu) << 16));
    g1[3] = (int)((td1 >> 16) | (tile0 << 16));
    g1[4] = (int)(tile1 | (tile2 << 16));
    g1[5] = (int)(unsigned int)s0;                // tensor_dim0_stride[31:0]
    g1[6] = (int)((unsigned int)(s0 >> 32) | ((unsigned int)(s1 & 0xFFFFu) << 16));
    g1[7] = (int)(unsigned int)(s1 >> 16);        // tensor_dim1_stride[47:16]
    i32x4 g2;
    g2[0] = (int)td2;                             // tensor_dim2
    g2[1] = 0; g2[2] = 0; g2[3] = 0;              // dim3 / dim2_stride / tile3
    i32x4 g3;
    g3[0] = 0; g3[1] = 0; g3[2] = 0; g3[3] = 0;
#if defined(__clang_major__) && __clang_major__ >= 23
    i32x8 z8;
    z8[0] = 0; z8[1] = 0; z8[2] = 0; z8[3] = 0;
    z8[4] = 0; z8[5] = 0; z8[6] = 0; z8[7] = 0;
    __builtin_amdgcn_tensor_load_to_lds(g0, g1, g2, g3, z8, 0);
#else
    __builtin_amdgcn_tensor_load_to_lds(g0, g1, g2, g3, 0);
#endif
}

__device__ __forceinline__ unsigned int lds_off(const void* p) {
    // generic LDS addresses carry the in-LDS byte offset in their low 32 bits
    return (unsigned int)(uintptr_t)p;
}
#endif // HAVE_TDM

// ---------------------------------------------------------------------------
// Generic fp32 WMMA GEMM:  Y[b][m][n] = sum_k Wt[m][k] * X[b][k][n] + bias[m]
// rows m < scaleRows additionally multiplied by scaleVal (q pre-scaling).
// Block: 128 threads = 4 waves; wave w computes the 16x16 tile at
// (m0, n0 + 16w). Requires K % 4 == 0, K <= 256, N % 64 == 0, M % 16 == 0.
// ---------------------------------------------------------------------------
__global__ __launch_bounds__(128)
void gemm16_wmma(const float* __restrict__ Wt, const float* __restrict__ X,
                 const float* __restrict__ bias, float* __restrict__ Y,
                 int M, int K, int N, long strideXb, long strideYb,
                 int scaleRows, float scaleVal)
{
    const int tid  = threadIdx.x;
    const int lane = tid & 31;
    const int wave = tid >> 5;
    const int m0   = blockIdx.x * 16;
    const int n0   = blockIdx.y * 64 + wave * 16;
    const int b    = blockIdx.z;

    __shared__ __align__(16) float sA[16 * 256];   // A tile [16][K]

    // cooperative vectorized load of the A tile (weights)
    const int K4 = K >> 2;
    for (int idx = tid; idx < 16 * K4; idx += 128) {
        int r = idx / K4, c = idx - r * K4;
        reinterpret_cast<v4f*>(sA)[r * K4 + c] =
            reinterpret_cast<const v4f*>(Wt)[(long)(m0 + r) * K4 + c];
    }
    __syncthreads();

    const float* Xb  = X + (long)b * strideXb;
    const int mrow   = lane & 15;
    const int koff   = (lane < 16) ? 0 : 2;
    const int hiOff  = (lane < 16) ? 0 : 8;

    v8f acc = {0.f, 0.f, 0.f, 0.f, 0.f, 0.f, 0.f, 0.f};
    for (int k0 = 0; k0 < K; k0 += 4) {
        const float* xc = Xb + (long)(k0 + koff) * N + n0 + mrow;
        if (k0 + 8 < K) __builtin_prefetch(xc + 8L * N, 0, 1);  // global_prefetch_b8
        v2f a, bb;
        a.x  = sA[mrow * K + k0 + koff];
        a.y  = sA[mrow * K + k0 + koff + 1];
        bb.x = xc[0];
        bb.y = xc[N];
        acc  = wmma_f32(a, bb, acc);
    }

    float* Yb = Y + (long)b * strideYb;
    #pragma unroll
    for (int r = 0; r < 8; ++r) {
        int   m = m0 + r + hiOff;
        float v = acc[r] + bias[m];
        if (m < scaleRows) v *= scaleVal;
        Yb[(long)m * N + n0 + mrow] = v;
    }
}

// ---------------------------------------------------------------------------
// Fused neighborhood attention (linear, no softmax):
//   out[q,d] = sum_{k in win(q)} (q . k + relbias) * v[k,d]
// One wave per (batch, head, row h, 16-query column tile).
// Key/value patch = 7 rows x 22 cols (union of the 16 clamped windows),
// staged into LDS by the Tensor Data Mover in [d][row][col] order.
// ---------------------------------------------------------------------------
__global__ __launch_bounds__(32)
void natt_wmma(const float* __restrict__ qkv,       // [B][3*256][3136]
               const float* __restrict__ relbias,   // [169][8]
               float* __restrict__ aout)            // [B][256][3136]
{
    const int lane = threadIdx.x;
    int w0 = blockIdx.x * 16; if (w0 > 40) w0 = 40;  // tiles 0,16,32,40
    const int h    = blockIdx.y;
    const int head = blockIdx.z & (NH - 1);
    const int b    = blockIdx.z >> 3;

    const int wh  = iclamp(h  - 3, 0, 49);   // window top row
    const int wc0 = iclamp(w0 - 3, 0, 49);   // leftmost union column

    const float* qf = qkv + ((long)(b * 768 +       head * HD)) * HWN;
    const float* kf = qkv + ((long)(b * 768 + 256 + head * HD)) * HWN;
    const float* vf = qkv + ((long)(b * 768 + 512 + head * HD)) * HWN;

    // TDM-natural layouts (x fastest): rows of 154 = 7*22 per d.
    __shared__ __align__(16) float sQt[HD * 16];          // [d][q]
    __shared__ __align__(16) float sKt[HD * KPOS + 32];   // [d][kpos] + slack
    __shared__ __align__(16) float sVt[HD * KPOS + 32];   // [d][kpos] + slack
    __shared__ float sS[16 * 161];                        // [q][kpos] padded
    __shared__ float sRB[169];

    // zero the slack tails: GEMM2 reads 0-weighted rows there, keep them finite
    sKt[HD * KPOS + lane] = 0.f;
    sVt[HD * KPOS + lane] = 0.f;
    for (int i = lane; i < 169; i += 32) sRB[i] = relbias[i * NH + head];

#if HAVE_TDM
    // Q: 2D tile 16(w) x 32(d); K/V: 3D tiles 22(w) x 7(h) x 32(d).
    // tensor_dim0 = remaining width -> TDM zero-fills the clamped right edge.
    tdm_load3d(lds_off(sQt), qf + h * WW2 + w0,
               (unsigned)(WW2 - w0), HD, 1,
               16, HD, 0,
               HWN, 0);
    tdm_load3d(lds_off(sKt), kf + wh * WW2 + wc0,
               (unsigned)(WW2 - wc0), (unsigned)(HH - wh), HD,
               22, 7, HD,
               WW2, HWN);
    tdm_load3d(lds_off(sVt), vf + wh * WW2 + wc0,
               (unsigned)(WW2 - wc0), (unsigned)(HH - wh), HD,
               22, 7, HD,
               WW2, HWN);
    __builtin_amdgcn_s_wait_tensorcnt(0);
#else
    for (int i = lane; i < HD * 16; i += 32) {
        int d = i >> 4, q = i & 15;
        sQt[i] = qf[(long)d * HWN + h * WW2 + (w0 + q)];
    }
    for (int i = lane; i < HD * KPOS; i += 32) {
        int d = i / KPOS, p = i - d * KPOS;
        int ir = p / 22, j = p - ir * 22;
        int kw = wc0 + j;
        float kk = 0.f, vv = 0.f;
        if (kw < WW2) {
            long off = (long)d * HWN + (wh + ir) * WW2 + kw;
            kk = kf[off];
            vv = vf[off];
        }
        sKt[i] = kk;
        sVt[i] = vv;
    }
#endif
    __syncthreads();

    const int mrow  = lane & 15;
    const int koff  = (lane < 16) ? 0 : 2;
    const int hiOff = (lane < 16) ? 0 : 8;

    // --- GEMM1: S[16 x 160] = Q[16 x 32] * K^T[32 x 160], mask + bias ---
    // (columns >= 154 compute garbage from row wrap; epilogue masks them)
    for (int t = 0; t < KPAD / 16; ++t) {
        v8f acc = {0.f, 0.f, 0.f, 0.f, 0.f, 0.f, 0.f, 0.f};
        #pragma unroll
        for (int k0 = 0; k0 < HD; k0 += 4) {
            v2f a, bb;
            a.x  = sQt[(k0 + koff)     * 16 + mrow];
            a.y  = sQt[(k0 + koff + 1) * 16 + mrow];
            bb.x = sKt[(k0 + koff)     * KPOS + t * 16 + mrow];
            bb.y = sKt[(k0 + koff + 1) * KPOS + t * 16 + mrow];
            acc  = wmma_f32(a, bb, acc);
        }
        const int kpos = t * 16 + mrow;
        const int irow = kpos / 22, j = kpos - irow * 22;
        #pragma unroll
        for (int r = 0; r < 8; ++r) {
            int qi  = r + hiOff;
            int wq  = w0 + qi;
            int wwq = iclamp(wq - 3, 0, 49);
            int dj  = j - (wwq - wc0);
            float val = 0.f;
            if (kpos < KPOS && dj >= 0 && dj <= 6) {
                int relh = (h - wh) + irow;
                int relw = (wq - wwq) + dj;
                val = acc[r] + sRB[relh * 13 + relw];
            }
            sS[qi * 161 + kpos] = val;
        }
    }
    __syncthreads();

    // --- GEMM2: O[16 x 32] = S[16 x 160] * V[160 x 32] ---
    // (k rows >= 154 carry S == 0, so wrapped/slack V rows contribute nothing)
    for (int t = 0; t < 2; ++t) {
        v8f acc = {0.f, 0.f, 0.f, 0.f, 0.f, 0.f, 0.f, 0.f};
        #pragma unroll
        for (int k0 = 0; k0 < KPAD; k0 += 4) {
            v2f a, bb;
            a.x  = sS[mrow * 161 + k0 + koff];
            a.y  = sS[mrow * 161 + k0 + koff + 1];
            bb.x = sVt[(t * 16 + mrow) * KPOS + k0 + koff];
            bb.y = sVt[(t * 16 + mrow) * KPOS + k0 + koff + 1];
            acc  = wmma_f32(a, bb, acc);
        }
        const int d = t * 16 + mrow;
        #pragma unroll
        for (int r = 0; r < 8; ++r) {
            int qi = r + hiOff;
            aout[((long)(b * 256 + head * HD + d)) * HWN + h * WW2 + (w0 + qi)] = acc[r];
        }
    }
}

// ---------------------------------------------------------------------------
extern "C" void kernel_launch(void* const* d_in, const int* in_sizes, int n_in,
                              void* d_out, int out_size, void* d_ws, size_t ws_size,
                              hipStream_t stream)
{
    const float* x       = (const float*)d_in[0];   // [2][256][56][56]
    const float* qkv_w   = (const float*)d_in[1];   // [768][256]
    const float* qkv_b   = (const float*)d_in[2];   // [768]
    const float* proj_w  = (const float*)d_in[3];   // [256][256]
    const float* proj_b  = (const float*)d_in[4];   // [256]
    const float* relbias = (const float*)d_in[5];   // [169][8]
    float*       out     = (float*)d_out;           // [2][256][56][56]

    float* qkv  = (float*)d_ws;                     // 2*768*3136 floats
    float* aout = qkv + 2L * 768 * HWN;             // 2*256*3136 floats

    const float qscale = 0.17677669529663687f;      // 32^-0.5

    // QKV projection (q rows pre-scaled)
    dim3 g1(768 / 16, HWN / 64, 2);
    gemm16_wmma<<<g1, 128, 0, stream>>>(qkv_w, x, qkv_b, qkv,
                                        768, 256, HWN,
                                        256L * HWN, 768L * HWN,
                                        256, qscale);

    // Fused neighborhood attention (TDM-staged tiles)
    dim3 g2(4, HH, 2 * NH);
    natt_wmma<<<g2, 32, 0, stream>>>(qkv, relbias, aout);

    // Output projection
    dim3 g3(256 / 16, HWN / 64, 2);
    gemm16_wmma<<<g3, 128, 0, stream>>>(proj_w, aout, proj_b, out,
                                        256, 256, HWN,
                                        256L * HWN, 256L * HWN,
                                        0, 1.0f);
}